// GCN_44736379355209
// MI455X (gfx1250) — compile-verified
//
#include <hip/hip_runtime.h>
#include <math.h>

// ---------------------------------------------------------------------------
// GCN (2-layer, PyG GCNConv semantics) for MI455X / gfx1250.
//   layer: out = D^-1/2 (A + I) D^-1/2 (X W) + b
// Dimensions: N=50000, din=dh=64, dout=16, E=1.6M, edge_index int64 [2,E].
//
// Strategy:
//  * GEMMs via V_WMMA_F32_16X16X4_F32; W staged in LDS (ds_load B fragments),
//    NCOLS templated so all strides are immediates.
//  * Edge aggregation: lane-coalesced float2 gathers + f32 atomics that
//    resolve in the 192MB L2 (whole ~32MB working set is L2-resident).
// ---------------------------------------------------------------------------

typedef __attribute__((ext_vector_type(2))) float v2f;
typedef __attribute__((ext_vector_type(8))) float v8f;

// ----------------------------- degree / norm -------------------------------

__global__ void gcn_deg_init(float* __restrict__ deg, int n) {
    int i = blockIdx.x * blockDim.x + threadIdx.x;
    if (i < n) deg[i] = 1.0f;  // self-loop contribution
}

__global__ void gcn_deg_accum(const long long* __restrict__ dst,
                              float* __restrict__ deg, long long e) {
    long long i = (long long)blockIdx.x * blockDim.x + threadIdx.x;
    if (i < e) atomicAdd(&deg[(int)dst[i]], 1.0f);
}

__global__ void gcn_deg_rsqrt(float* __restrict__ deg, int n) {
    int i = blockIdx.x * blockDim.x + threadIdx.x;
    if (i < n) {
        float d = deg[i];
        deg[i] = (d > 0.0f) ? rsqrtf(d) : 0.0f;
    }
}

// ------------------------------ WMMA GEMM ----------------------------------
// H[n, NCOLS] = X[n, 64] @ W[64, NCOLS].  One wave per 16x16 output tile,
// 8 waves (256 threads) per block, W cached in LDS for the whole block.
//
// A 16x4 f32 layout: lanes 0-15 -> M=lane, K=k+{0,1}; lanes 16-31 -> K=k+{2,3}.
// B 4x16 f32 layout: K-row striped across VGPRs, N = lane&15.
// C 16x16 f32 layout: vgpr r, lane l -> row = r + 8*(l>=16), col = l&15.

template <int NCOLS>
__global__ void gcn_gemm_wmma(const float* __restrict__ X,
                              const float* __restrict__ W,
                              float* __restrict__ H, int n) {
    constexpr int CTILES = NCOLS / 16;
    __shared__ float sW[64 * NCOLS];

    // cooperative stage of W into LDS (16KB / 4KB)
    for (int i = threadIdx.x; i < 64 * NCOLS; i += blockDim.x)
        sW[i] = W[i];
    __syncthreads();

    const int wave    = blockIdx.x * (blockDim.x >> 5) + (threadIdx.x >> 5);
    const int lane    = threadIdx.x & 31;
    const int rowTile = wave / CTILES;   // CTILES is a power of two -> shift
    const int colTile = wave - rowTile * CTILES;
    const int m0 = rowTile << 4;
    if (m0 >= n) return;  // wave-uniform exit: EXEC all-1s inside

    const int n0  = colTile << 4;
    const int sub = lane & 15;           // A row index / B,C column index
    const int khi = (lane >> 4) << 1;    // 0 or 2: K sub-offset per half-wave

    const float* __restrict__ xrow = X + (size_t)(m0 + sub) * 64;

    // prefetch the next row tile of X (global_prefetch_b8)
    if (m0 + 16 < n)
        __builtin_prefetch(X + (size_t)(m0 + 16 + sub) * 64, 0, 0);

    const float* __restrict__ wcol = sW + n0 + sub;

    v8f acc = {};
    #pragma unroll
    for (int k = 0; k < 64; k += 4) {
        // A fragment: two contiguous f32 -> one global_load_b64
        v2f a = *(const v2f*)(xrow + k + khi);
        // B fragment from LDS: two ds_load_b32 (stride NCOLS is immediate)
        v2f b;
        b.x = wcol[(k + khi + 0) * NCOLS];
        b.y = wcol[(k + khi + 1) * NCOLS];
        // D = A(16x4) * B(4x16) + C   -> v_wmma_f32_16x16x4_f32
        acc = __builtin_amdgcn_wmma_f32_16x16x4_f32(
            /*neg_a=*/false, a, /*neg_b=*/false, b,
            /*c_mod=*/(short)0, acc, /*reuse_a=*/false, /*reuse_b=*/false);
    }

    const int rbase = (lane >> 4) << 3;  // 0 or 8
    float* __restrict__ hout = H + (size_t)m0 * NCOLS + n0 + sub;
    #pragma unroll
    for (int r = 0; r < 8; ++r)
        hout[(size_t)(rbase + r) * NCOLS] = acc[r];
}

// ------------------------- aggregation kernels -----------------------------

// seed with self-loop term: out[i,:] = dinv[i]^2 * h[i,:]
template <int NCOLS>
__global__ void gcn_selfloop_seed(const float* __restrict__ h,
                                  const float* __restrict__ dinv,
                                  float* __restrict__ out, long long total) {
    long long t = (long long)blockIdx.x * blockDim.x + threadIdx.x;
    if (t < total) {
        int i = (int)(t / NCOLS);
        float di = dinv[i];
        out[t] = di * di * h[t];
    }
}

// scatter-add of edge messages. LPE lanes cooperate on one edge; each lane
// handles VEC = FEATS/LPE contiguous features (float2 gather when VEC==2).
template <int FEATS, int LPE>
__global__ void gcn_edge_scatter(const long long* __restrict__ src,
                                 const long long* __restrict__ dst,
                                 const float* __restrict__ dinv,
                                 const float* __restrict__ h,
                                 float* __restrict__ out, long long e) {
    constexpr int VEC = FEATS / LPE;
    long long t = (long long)blockIdx.x * blockDim.x + threadIdx.x;
    long long eidx = t / LPE;
    int lane = (int)(t - eidx * LPE);
    if (eidx >= e) return;
    int s = (int)src[eidx];
    int d = (int)dst[eidx];
    float norm = dinv[s] * dinv[d];
    const float* __restrict__ hs = h + (size_t)s * FEATS + lane * VEC;
    float* __restrict__ od = out + (size_t)d * FEATS + lane * VEC;
    if constexpr (VEC == 2) {
        v2f hv = *(const v2f*)hs;    // coalesced global_load_b64 across wave
        atomicAdd(&od[0], norm * hv.x);
        atomicAdd(&od[1], norm * hv.y);
    } else {
        atomicAdd(&od[0], norm * hs[0]);
    }
}

template <int NCOLS>
__global__ void gcn_bias_relu(float* __restrict__ out,
                              const float* __restrict__ b, long long total) {
    long long t = (long long)blockIdx.x * blockDim.x + threadIdx.x;
    if (t < total) {
        float v = out[t] + b[(int)(t & (NCOLS - 1))];
        out[t] = fmaxf(v, 0.0f);
    }
}

// final: add bias, then log_softmax over 16 classes; one thread per node row
__global__ void gcn_bias_logsoftmax16(const float* __restrict__ agg,
                                      const float* __restrict__ b,
                                      float* __restrict__ out, int n) {
    int i = blockIdx.x * blockDim.x + threadIdx.x;
    if (i >= n) return;
    const float* __restrict__ row = agg + (size_t)i * 16;
    float v[16];
    float m = -INFINITY;
    #pragma unroll
    for (int j = 0; j < 16; ++j) {
        v[j] = row[j] + b[j];
        m = fmaxf(m, v[j]);
    }
    float s = 0.0f;
    #pragma unroll
    for (int j = 0; j < 16; ++j) s += expf(v[j] - m);
    float ls = logf(s);
    float* __restrict__ orow = out + (size_t)i * 16;
    #pragma unroll
    for (int j = 0; j < 16; ++j) orow[j] = v[j] - m - ls;
}

// ------------------------------- launcher ----------------------------------

extern "C" void kernel_launch(void* const* d_in, const int* in_sizes, int n_in,
                              void* d_out, int out_size, void* d_ws, size_t ws_size,
                              hipStream_t stream) {
    (void)n_in; (void)out_size; (void)ws_size;

    const float*     x    = (const float*)d_in[0];      // [N, 64]
    const long long* eidx = (const long long*)d_in[1];  // [2, E] int64
    const float*     W1   = (const float*)d_in[2];      // [64, 64]
    const float*     b1   = (const float*)d_in[3];      // [64]
    const float*     W2   = (const float*)d_in[4];      // [64, 16]
    const float*     b2   = (const float*)d_in[5];      // [16]
    float*           out  = (float*)d_out;              // [N, 16]

    const int       n = in_sizes[0] / 64;               // 50000
    const long long E = (long long)in_sizes[1] / 2;     // 1.6M
    const long long* src = eidx;
    const long long* dst = eidx + E;

    // carve workspace
    float* ws   = (float*)d_ws;
    float* dinv = ws;                       // [N]      (deg -> dinv in place)
    float* h1   = dinv + n;                 // [N, 64]
    float* agg1 = h1 + (size_t)n * 64;      // [N, 64]
    float* h2   = agg1 + (size_t)n * 64;    // [N, 16]
    float* agg2 = h2 + (size_t)n * 16;      // [N, 16]

    const int B = 256;
    auto blocks = [](long long work, int blk) -> unsigned {
        return (unsigned)((work + blk - 1) / blk);
    };

    // 1) degrees (with self-loop) -> dinv
    gcn_deg_init<<<blocks(n, B), B, 0, stream>>>(dinv, n);
    gcn_deg_accum<<<blocks(E, B), B, 0, stream>>>(dst, dinv, E);
    gcn_deg_rsqrt<<<blocks(n, B), B, 0, stream>>>(dinv, n);

    // 2) layer 1 GEMM: h1 = x @ W1   (3125 row tiles x 4 col tiles, 8 waves/blk)
    {
        long long waves = (long long)((n + 15) / 16) * 4;
        gcn_gemm_wmma<64><<<blocks(waves * 32, B), B, 0, stream>>>(x, W1, h1, n);
    }

    // 3) layer 1 aggregation
    gcn_selfloop_seed<64><<<blocks((long long)n * 64, B), B, 0, stream>>>(
        h1, dinv, agg1, (long long)n * 64);
    gcn_edge_scatter<64, 32><<<blocks(E * 32, B), B, 0, stream>>>(
        src, dst, dinv, h1, agg1, E);
    gcn_bias_relu<64><<<blocks((long long)n * 64, B), B, 0, stream>>>(
        agg1, b1, (long long)n * 64);

    // 4) layer 2 GEMM: h2 = relu(agg1) @ W2   (3125 row tiles x 1 col tile)
    {
        long long waves = (long long)((n + 15) / 16);
        gcn_gemm_wmma<16><<<blocks(waves * 32, B), B, 0, stream>>>(agg1, W2, h2, n);
    }

    // 5) layer 2 aggregation + bias + log_softmax
    gcn_selfloop_seed<16><<<blocks((long long)n * 16, B), B, 0, stream>>>(
        h2, dinv, agg2, (long long)n * 16);
    gcn_edge_scatter<16, 16><<<blocks(E * 16, B), B, 0, stream>>>(
        src, dst, dinv, h2, agg2, E);
    gcn_bias_logsoftmax16<<<blocks(n, B), B, 0, stream>>>(agg2, b2, out, n);
}